// Loc2Cluster_41188736369204
// MI455X (gfx1250) — compile-verified
//
#include <hip/hip_runtime.h>
#include <cstdint>
#include <cstddef>

// ---------------------------------------------------------------------------
// Loc2Cluster segment-max + concat for MI455X (gfx1250, wave32)
//   out[c, 0:256]   = x_clusters[c]
//   out[c, 256:512] = max over {x_locs[edge_src[e]] : edge_dst[e]==c}, or 0 if none
// Pipeline: zero -> histogram -> prefix scan -> CSR fill -> wave-per-cluster max
// ---------------------------------------------------------------------------

#if defined(__HIP_DEVICE_COMPILE__)
#if __has_builtin(__builtin_amdgcn_global_load_async_to_lds_b128) && \
    __has_builtin(__builtin_amdgcn_global_store_async_from_lds_b128)
#define L2C_ASYNC 1
#endif
#endif

// Native vector type: required for __builtin_nontemporal_load and gives
// __builtin_elementwise_max -> v_max_num_f32.
typedef float l2c_f4 __attribute__((ext_vector_type(4)));

#if defined(L2C_ASYNC)
// b128 async builtins take (int4-vector AS1 ptr, int4-vector AS3 ptr, imm, imm).
typedef int l2c_v4i __attribute__((vector_size(16)));
typedef __attribute__((address_space(1))) l2c_v4i* l2c_gptr;  // global int4*
typedef __attribute__((address_space(3))) l2c_v4i* l2c_lptr;  // LDS int4*

__device__ __forceinline__ l2c_gptr l2c_to_global(const void* p) {
  return (l2c_gptr)(void*)p;  // strip const, addrspacecast generic -> AS1
}
__device__ __forceinline__ l2c_lptr l2c_to_lds(void* p) {
  return (l2c_lptr)p;         // addrspacecast generic -> AS3
}
#endif

__device__ __forceinline__ l2c_f4 l2c_ntload(const l2c_f4* p) {
  return __builtin_nontemporal_load(p);  // global_load_b128 ... th:TH_LOAD_NT
}

// --- hot kernel first so the disasm snippet shows it -------------------------
// One wave32 per cluster. D == 256: each lane owns 8 features as two float4s.
// All loads/stores are B128 and coalesced (512B per wave instruction).
// x_locs rows are streamed non-temporally (256MB, read-once, > 192MB L2).
// The x_clusters->out copy goes through the async global<->LDS engine
// (ASYNCcnt), overlapping with the v_max_num_f32 accumulation loop.
__global__ void __launch_bounds__(256) l2c_aggcat_kernel(
    const float* __restrict__ x_locs, const float* __restrict__ x_clusters,
    const int* __restrict__ offsets, const int* __restrict__ csr,
    float* __restrict__ out, int C) {
  const int lane = threadIdx.x & 31;
  const int wave_in_blk = threadIdx.x >> 5;
  const int c = blockIdx.x * (blockDim.x >> 5) + wave_in_blk;
  if (c >= C) return;  // wave-uniform exit

  const l2c_f4* __restrict__ crow = (const l2c_f4*)(x_clusters + (size_t)c * 256u);
  l2c_f4* __restrict__ orow = (l2c_f4*)(out + (size_t)c * 512u);

#if defined(L2C_ASYNC)
  __shared__ __align__(16) float stage[8][256];
  float* lds = &stage[wave_in_blk][0];
  // Kick off the x_clusters row copy into LDS; it completes while we do the max.
  __builtin_amdgcn_global_load_async_to_lds_b128(
      l2c_to_global(crow + lane),      l2c_to_lds(lds + lane * 4),       0, 0);
  __builtin_amdgcn_global_load_async_to_lds_b128(
      l2c_to_global(crow + 32 + lane), l2c_to_lds(lds + 128 + lane * 4), 0, 0);
#else
  const l2c_f4 c0 = crow[lane];
  const l2c_f4 c1 = crow[32 + lane];
#endif

  const int begin = offsets[c];
  const int end   = offsets[c + 1];
  const float ninf = -__builtin_inff();
  l2c_f4 a0 = {ninf, ninf, ninf, ninf};
  l2c_f4 a1 = a0;

  // 2-row unrolled gather-max: 4 outstanding B128 loads per iteration.
  int i = begin;
  for (; i + 1 < end; i += 2) {
    const l2c_f4* __restrict__ r0 = (const l2c_f4*)(x_locs + (size_t)csr[i] * 256u);
    const l2c_f4* __restrict__ r1 = (const l2c_f4*)(x_locs + (size_t)csr[i + 1] * 256u);
    if (i + 2 < end) {  // prefetch the next pair's first row (global_prefetch_b8)
      const char* nxt = (const char*)(x_locs + (size_t)csr[i + 2] * 256u);
      __builtin_prefetch(nxt + (size_t)(lane & 15) * 64u, 0, 0);
    }
    l2c_f4 v0 = l2c_ntload(r0 + lane);
    l2c_f4 v1 = l2c_ntload(r0 + 32 + lane);
    l2c_f4 w0 = l2c_ntload(r1 + lane);
    l2c_f4 w1 = l2c_ntload(r1 + 32 + lane);
    a0 = __builtin_elementwise_max(__builtin_elementwise_max(a0, v0), w0);
    a1 = __builtin_elementwise_max(__builtin_elementwise_max(a1, v1), w1);
  }
  if (i < end) {  // tail row
    const l2c_f4* __restrict__ r0 = (const l2c_f4*)(x_locs + (size_t)csr[i] * 256u);
    a0 = __builtin_elementwise_max(a0, l2c_ntload(r0 + lane));
    a1 = __builtin_elementwise_max(a1, l2c_ntload(r0 + 32 + lane));
  }

  if (begin == end) {  // empty cluster: PyG scatter-max semantics -> 0
    a0 = (l2c_f4){0.f, 0.f, 0.f, 0.f};
    a1 = a0;
  }
  orow[64 + lane] = a0;
  orow[96 + lane] = a1;

#if defined(L2C_ASYNC)
  // Async LDS writes (loads) vs LDS reads (stores) are not ordered: fence them.
#if __has_builtin(__builtin_amdgcn_s_wait_asynccnt)
  __builtin_amdgcn_s_wait_asynccnt(0);
#else
  asm volatile("s_wait_asynccnt 0" ::: "memory");
#endif
  __builtin_amdgcn_global_store_async_from_lds_b128(
      l2c_to_global(orow + lane),      l2c_to_lds(lds + lane * 4),       0, 0);
  __builtin_amdgcn_global_store_async_from_lds_b128(
      l2c_to_global(orow + 32 + lane), l2c_to_lds(lds + 128 + lane * 4), 0, 0);
  // S_ENDPGM performs an implicit wait-idle, so outstanding async stores drain.
#else
  orow[lane]      = c0;
  orow[32 + lane] = c1;
#endif
}

// --- pass 0: zero counts + cursors (ws state must be reset on every call) ---
__global__ void l2c_zero_kernel(int* __restrict__ p, int n) {
  int i = blockIdx.x * blockDim.x + threadIdx.x;
  if (i < n) p[i] = 0;
}

// --- pass 1: per-cluster edge histogram --------------------------------------
__global__ void l2c_hist_kernel(const int* __restrict__ edge_dst,
                                int* __restrict__ counts, int E) {
  int e = blockIdx.x * blockDim.x + threadIdx.x;
  if (e < E) atomicAdd(&counts[edge_dst[e]], 1);
}

// --- pass 2: exclusive prefix scan over C counts (single 1024-thread block) --
__global__ void __launch_bounds__(1024) l2c_scan_kernel(
    const int* __restrict__ counts, int* __restrict__ offsets, int C) {
  __shared__ int sdata[1024];
  const int t = threadIdx.x;
  const int items = (C + 1023) >> 10;
  const int base = t * items;

  int sum = 0;
  for (int i = 0; i < items; ++i) {
    int idx = base + i;
    if (idx < C) sum += counts[idx];
  }
  sdata[t] = sum;
  __syncthreads();

  int acc = sum;  // Hillis-Steele inclusive scan of per-thread sums
  for (int off = 1; off < 1024; off <<= 1) {
    int v = (t >= off) ? sdata[t - off] : 0;
    __syncthreads();
    acc += v;
    sdata[t] = acc;
    __syncthreads();
  }

  int run = acc - sum;  // exclusive prefix for this thread's chunk
  for (int i = 0; i < items; ++i) {
    int idx = base + i;
    if (idx < C) { offsets[idx] = run; run += counts[idx]; }
  }
  if (t == 1023) offsets[C] = acc;  // total = E
}

// --- pass 3: scatter loc indices into CSR order ------------------------------
__global__ void l2c_fill_kernel(const int* __restrict__ edge_src,
                                const int* __restrict__ edge_dst,
                                const int* __restrict__ offsets,
                                int* __restrict__ cursor,
                                int* __restrict__ csr, int E) {
  int e = blockIdx.x * blockDim.x + threadIdx.x;
  if (e < E) {
    int d = edge_dst[e];
    int pos = offsets[d] + atomicAdd(&cursor[d], 1);
    csr[pos] = edge_src[e];
  }
}

// ---------------------------------------------------------------------------
extern "C" void kernel_launch(void* const* d_in, const int* in_sizes, int n_in,
                              void* d_out, int out_size, void* d_ws, size_t ws_size,
                              hipStream_t stream) {
  (void)n_in; (void)out_size; (void)ws_size;

  const float* x_locs     = (const float*)d_in[0];
  const float* x_clusters = (const float*)d_in[1];
  const int*   edge_src   = (const int*)d_in[2];
  const int*   edge_dst   = (const int*)d_in[3];
  float* out = (float*)d_out;

  const int E = in_sizes[2];            // 262144 edges (== locs)
  const int D = in_sizes[0] / E;        // 256
  const int C = in_sizes[1] / D;        // 32768 clusters

  // Workspace layout (~1.5 MB): counts[C] | cursor[C] | offsets[C+1] | csr[E]
  int* counts  = (int*)d_ws;
  int* cursor  = counts + C;
  int* offsets = cursor + C;
  int* csr     = offsets + (C + 1);

  {
    int n = 2 * C;
    l2c_zero_kernel<<<(n + 255) / 256, 256, 0, stream>>>(counts, n);
  }
  l2c_hist_kernel<<<(E + 255) / 256, 256, 0, stream>>>(edge_dst, counts, E);
  l2c_scan_kernel<<<1, 1024, 0, stream>>>(counts, offsets, C);
  l2c_fill_kernel<<<(E + 255) / 256, 256, 0, stream>>>(edge_src, edge_dst,
                                                       offsets, cursor, csr, E);
  l2c_aggcat_kernel<<<(C + 7) / 8, 256, 0, stream>>>(x_locs, x_clusters,
                                                     offsets, csr, out, C);
}